// Net_5729486373069
// MI455X (gfx1250) — compile-verified
//
#include <hip/hip_runtime.h>
#include <math.h>

typedef float v2f __attribute__((ext_vector_type(2)));
typedef float v8f __attribute__((ext_vector_type(8)));

#define NN 100000          // nodes
#define NE 1600000         // edges
#define FIN 256
#define H1 8
#define D1 8
#define HD1 64             // H1*D1
#define C2 16              // classes
#define NEG_SLOPE 0.2f
#define FLIP_NEG_INF 0x007FFFFFu   // fflip(-inf)

// ---------- helpers ----------
__device__ __forceinline__ unsigned fflip(float f) {
    unsigned u = __float_as_uint(f);
    return (u & 0x80000000u) ? ~u : (u | 0x80000000u);
}
__device__ __forceinline__ float funflip(unsigned u) {
    return __uint_as_float((u & 0x80000000u) ? (u & 0x7fffffffu) : ~u);
}
__device__ __forceinline__ float lrelu(float v) { return v > 0.0f ? v : NEG_SLOPE * v; }

// ---------- init ----------
__global__ void fill_f32(float* p, float v, int n) {
    int i = blockIdx.x * blockDim.x + threadIdx.x;
    if (i < n) p[i] = v;
}
__global__ void fill_u32(unsigned* p, unsigned v, int n) {
    int i = blockIdx.x * blockDim.x + threadIdx.x;
    if (i < n) p[i] = v;
}

// ---------- fp32 WMMA GEMM: C[M, NT*16] = A[M,K] @ B[K, NT*16], 16|M, 4|K ----------
// One wave per 16-row strip covering ALL NT column tiles: the A fragment is
// loaded once per k-step and reused across NT v_wmma_f32_16x16x4_f32 issues.
template <int NT>
__global__ void gemm_wmma_f32(const float* __restrict__ A, const float* __restrict__ B,
                              float* __restrict__ C, int M, int K) {
    const int N = NT * 16;
    int wave = (int)((blockIdx.x * blockDim.x + threadIdx.x) >> 5);
    int lane = threadIdx.x & 31;
    int tilesM = M >> 4;
    if (wave >= tilesM) return;             // wave-uniform; EXEC all-1 in live waves

    int r  = lane & 15;                     // A: row index / B: col index
    int kh = (lane >> 4) << 1;              // 0 or 2 (K sub-slot per lane half)
    const float* Arow = A + (size_t)(wave * 16 + r) * K;

    v8f acc[NT];
#pragma unroll
    for (int j = 0; j < NT; ++j) acc[j] = (v8f){};

#pragma unroll 4
    for (int k = 0; k < K; k += 4) {
        v2f a;
        a.x = Arow[k + kh];
        a.y = Arow[k + kh + 1];
#pragma unroll
        for (int j = 0; j < NT; ++j) {
            const float* Bc = B + j * 16 + r;
            v2f b;
            b.x = Bc[(size_t)(k + kh) * N];
            b.y = Bc[(size_t)(k + kh + 1) * N];
            acc[j] = __builtin_amdgcn_wmma_f32_16x16x4_f32(
                /*neg_a=*/false, a, /*neg_b=*/false, b,
                /*c_mod=*/(short)0, acc[j], /*reuse_a=*/false, /*reuse_b=*/false);
        }
    }
    // C/D layout: VGPR i -> M = i (lanes 0-15) or i+8 (lanes 16-31), N = lane&15
    int cr = wave * 16 + ((lane >> 4) << 3);
#pragma unroll
    for (int j = 0; j < NT; ++j) {
        int cc = j * 16 + r;
#pragma unroll
        for (int i = 0; i < 8; ++i) C[(size_t)(cr + i) * N + cc] = acc[j][i];
    }
}

// ---------- per-node attention projections: s/t[n,h] = <h[n,h,:], a_src/dst[h,:]> ----------
__global__ void node_st(const float* __restrict__ h, const float* __restrict__ a_src,
                        const float* __restrict__ a_dst, float* __restrict__ s,
                        float* __restrict__ t, int nNodes, int H, int D) {
    int i = blockIdx.x * blockDim.x + threadIdx.x;
    if (i >= nNodes * H) return;
    int n = i / H, hh = i - n * H;
    const float* hp = h + (size_t)n * H * D + hh * D;
    const float* as = a_src + hh * D;
    const float* ad = a_dst + hh * D;
    float ss = 0.0f, tt = 0.0f;
    for (int d = 0; d < D; ++d) { float v = hp[d]; ss += v * as[d]; tt += v * ad[d]; }
    s[i] = ss; t[i] = tt;
}

// ---------- softmax pass 1: segment max over dst (order-preserving uint atomicMax) ----------
__global__ void edge_attn_max(const int* __restrict__ src, const int* __restrict__ dst,
                              const float* __restrict__ s, const float* __restrict__ t,
                              unsigned* __restrict__ mflip, int E, int H) {
    int e = blockIdx.x * blockDim.x + threadIdx.x;
    if (e >= E) return;
    int a = src[e], b = dst[e];
    for (int h = 0; h < H; ++h) {
        float v = lrelu(s[a * H + h] + t[b * H + h]);
        atomicMax(&mflip[b * H + h], fflip(v));
    }
}

// decode flipped max in place; -inf (no incoming edges) -> 0, matching reference
__global__ void m_finalize(unsigned* m, int n) {
    int i = blockIdx.x * blockDim.x + threadIdx.x;
    if (i >= n) return;
    float f = funflip(m[i]);
    if (!isfinite(f)) f = 0.0f;
    m[i] = __float_as_uint(f);
}

// ---------- softmax pass 2: denom[dst,h] = sum exp(e - m[dst,h]) ----------
__global__ void edge_denom(const int* __restrict__ src, const int* __restrict__ dst,
                           const float* __restrict__ s, const float* __restrict__ t,
                           const float* __restrict__ m, float* __restrict__ denom,
                           int E, int H) {
    int e = blockIdx.x * blockDim.x + threadIdx.x;
    if (e >= E) return;
    int a = src[e], b = dst[e];
    for (int h = 0; h < H; ++h) {
        float v = lrelu(s[a * H + h] + t[b * H + h]);
        atomicAdd(&denom[b * H + h], __expf(v - m[b * H + h]));
    }
}

// ---------- softmax pass 3: out[dst,h,:] += alpha * h[src,h,:]  (thread = edge*H + h) ----------
__global__ void edge_scatter(const int* __restrict__ src, const int* __restrict__ dst,
                             const float* __restrict__ s, const float* __restrict__ t,
                             const float* __restrict__ m, const float* __restrict__ denom,
                             const float* __restrict__ h, float* __restrict__ out,
                             int E, int H, int D) {
    int i = blockIdx.x * blockDim.x + threadIdx.x;
    if (i >= E * H) return;
    int e = i / H, hh = i - e * H;
    int a = src[e], b = dst[e];
    float v = lrelu(s[a * H + hh] + t[b * H + hh]);
    float alpha = __expf(v - m[b * H + hh]) / (denom[b * H + hh] + 1e-16f);
    const float* hs = h + (size_t)a * H * D + hh * D;
    float* ob = out + (size_t)b * H * D + hh * D;
    for (int d = 0; d < D; ++d) atomicAdd(&ob[d], alpha * hs[d]);
}

// ---------- elementwise ----------
__global__ void elu_ip(float* p, int n) {
    int i = blockIdx.x * blockDim.x + threadIdx.x;
    if (i < n) { float v = p[i]; p[i] = v > 0.0f ? v : (__expf(v) - 1.0f); }
}

__global__ void log_softmax_ip(float* p, int nNodes, int C) {
    int i = blockIdx.x * blockDim.x + threadIdx.x;
    if (i >= nNodes) return;
    float* r = p + (size_t)i * C;
    float mx = r[0];
    for (int c = 1; c < C; ++c) mx = fmaxf(mx, r[c]);
    float sum = 0.0f;
    for (int c = 0; c < C; ++c) sum += __expf(r[c] - mx);
    float l = __logf(sum);
    for (int c = 0; c < C; ++c) r[c] = r[c] - mx - l;
}

// ---------- launcher ----------
extern "C" void kernel_launch(void* const* d_in, const int* in_sizes, int n_in,
                              void* d_out, int out_size, void* d_ws, size_t ws_size,
                              hipStream_t stream) {
    const float* x    = (const float*)d_in[0];
    const int*   edge = (const int*)d_in[1];         // int32 (JAX default x64-off)
    const float* W1   = (const float*)d_in[2];
    const float* a1s  = (const float*)d_in[3];
    const float* a1d  = (const float*)d_in[4];
    const float* W2   = (const float*)d_in[5];
    const float* a2s  = (const float*)d_in[6];
    const float* a2d  = (const float*)d_in[7];
    const int* src = edge;            // row 0
    const int* dst = edge + NE;       // row 1
    float* out = (float*)d_out;       // [NN, 16]; also layer-2 accumulator

    // workspace layout (floats); layer-2 buffers overlay dead layer-1 buffers
    float*    ws   = (float*)d_ws;
    float*    h1   = ws;                                  // NN*64
    float*    out1 = ws + (size_t)NN * HD1;               // NN*64 (-> elu -> GEMM2 input)
    float*    s1   = out1 + (size_t)NN * HD1;             // NN*8
    float*    t1   = s1 + NN * H1;                        // NN*8
    unsigned* m1   = (unsigned*)(t1 + NN * H1);           // NN*8
    float*    den1 = (float*)m1 + NN * H1;                // NN*8
    float*    h2   = s1;                                  // NN*16 (overlays s1+t1)
    float*    s2   = den1 + NN * H1;                      // NN
    float*    t2   = s2 + NN;                             // NN
    unsigned* m2   = (unsigned*)(t2 + NN);                // NN
    float*    den2 = (float*)m2 + NN;                     // NN

    const int TB = 256;                                   // 8 waves / block
    #define GRID(n) (((n) + TB - 1) / TB)
    const int WAVES_M = NN / 16;                          // 6250 row-strip waves
    const int GEMM_BLOCKS = (WAVES_M + 7) / 8;

    // ===== layer 1 =====
    gemm_wmma_f32<4><<<GEMM_BLOCKS, TB, 0, stream>>>(x, W1, h1, NN, FIN);
    node_st<<<GRID(NN * H1), TB, 0, stream>>>(h1, a1s, a1d, s1, t1, NN, H1, D1);
    fill_u32<<<GRID(NN * H1), TB, 0, stream>>>(m1, FLIP_NEG_INF, NN * H1);
    fill_f32<<<GRID(NN * H1), TB, 0, stream>>>(den1, 0.0f, NN * H1);
    fill_f32<<<GRID(NN * HD1), TB, 0, stream>>>(out1, 0.0f, NN * HD1);
    edge_attn_max<<<GRID(NE), TB, 0, stream>>>(src, dst, s1, t1, m1, NE, H1);
    m_finalize<<<GRID(NN * H1), TB, 0, stream>>>(m1, NN * H1);
    edge_denom<<<GRID(NE), TB, 0, stream>>>(src, dst, s1, t1, (const float*)m1, den1, NE, H1);
    edge_scatter<<<GRID(NE * H1), TB, 0, stream>>>(src, dst, s1, t1, (const float*)m1, den1,
                                                   h1, out1, NE, H1, D1);
    elu_ip<<<GRID(NN * HD1), TB, 0, stream>>>(out1, NN * HD1);

    // ===== layer 2 =====
    gemm_wmma_f32<1><<<GEMM_BLOCKS, TB, 0, stream>>>(out1, W2, h2, NN, HD1);
    node_st<<<GRID(NN), TB, 0, stream>>>(h2, a2s, a2d, s2, t2, NN, 1, C2);
    fill_u32<<<GRID(NN), TB, 0, stream>>>(m2, FLIP_NEG_INF, NN);
    fill_f32<<<GRID(NN), TB, 0, stream>>>(den2, 0.0f, NN);
    fill_f32<<<GRID(NN * C2), TB, 0, stream>>>(out, 0.0f, NN * C2);
    edge_attn_max<<<GRID(NE), TB, 0, stream>>>(src, dst, s2, t2, m2, NE, 1);
    m_finalize<<<GRID(NN), TB, 0, stream>>>(m2, NN);
    edge_denom<<<GRID(NE), TB, 0, stream>>>(src, dst, s2, t2, (const float*)m2, den2, NE, 1);
    edge_scatter<<<GRID(NE), TB, 0, stream>>>(src, dst, s2, t2, (const float*)m2, den2,
                                              h2, out, NE, 1, C2);
    log_softmax_ip<<<GRID(NN), TB, 0, stream>>>(out, NN, C2);
    #undef GRID
}